// RelationalEncoder_88837103551531
// MI455X (gfx1250) — compile-verified
//
#include <hip/hip_runtime.h>
#include <stdint.h>

#define N_AGENT 50000
#define N_SRC   50000
#define E_EDGE  1000000
#define H_HEADS 4
#define HID     128
#define F_AGENT 32
#define F_SRC   16
#define NEG_SLOPE 0.2f

typedef __attribute__((ext_vector_type(2))) float    v2f;
typedef __attribute__((ext_vector_type(8))) float    v8f;
typedef __attribute__((ext_vector_type(4))) unsigned v4u;
typedef __attribute__((ext_vector_type(8))) int      v8i;
typedef __attribute__((ext_vector_type(4))) int      v4i;

typedef __attribute__((address_space(3))) float lds_f32;

// ---------------------------------------------------------------------------
// fp32 WMMA GEMM with TDM-staged weights:
//   out[m][n] = act( sum_k A[m][k] * W[n][k] + bias[n] ),  N fixed at 128.
// One TENSOR_LOAD_TO_LDS DMA stages W (128 x K fp32, row-major) into LDS per
// workgroup; 8 waves/block each compute a 16-row x 128-col strip. Per k-step
// all 8 LDS B-fragments are preloaded (clauseable ds_load_2addr batch) before
// the 8 back-to-back V_WMMA_F32_16X16X4_F32 ops, hiding LDS latency instead
// of a per-WMMA s_wait_dscnt 0x0 stall.
// ---------------------------------------------------------------------------
__global__ __launch_bounds__(256)
void wmma_gemm_tdm_kernel(const float* __restrict__ A, int lda,
                          const float* __restrict__ W,
                          const float* __restrict__ bias,
                          float* __restrict__ out, int ldo,
                          int M, int K, int doRelu) {
    extern __shared__ float ldsW[];            // 128 * K floats

    // ---- stage W into LDS with the Tensor Data Mover (one op per block) ----
    if (threadIdx.x == 0) {
        const unsigned long long gaddr = (unsigned long long)(uintptr_t)W;
        const unsigned ldsOff = (unsigned)(unsigned long long)(lds_f32*)ldsW;
        const unsigned uK = (unsigned)K;

        v4u g0;                                   // D# group 0 (ISA 8.3)
        g0.x = 1u;                                // count=1, is_restore=0
        g0.y = ldsOff;                            // lds_addr (bytes)
        g0.z = (unsigned)(gaddr & 0xFFFFFFFFull); // global_addr[31:0]
        g0.w = (unsigned)((gaddr >> 32) & 0x1FFFFFFull) | (2u << 30); // [56:32] | type=2

        v8i g1;                                   // D# group 1 (ISA 8.4)
        g1[0] = (int)(2u << 16);                  // wg_mask=0, data_size=2 (4 bytes)
        g1[1] = (int)((uK & 0xFFFFu) << 16);      // tensor_dim0[15:0] @ bits 63:48
        g1[2] = (int)(((uK >> 16) & 0xFFFFu) | (128u << 16)); // dim0 hi | tensor_dim1=128
        g1[3] = (int)((uK & 0xFFFFu) << 16);      // dim1 hi=0 | tile_dim0=K
        g1[4] = (int)(128u);                      // tile_dim1=128, tile_dim2=0
        g1[5] = (int)uK;                          // tensor_dim0_stride[31:0] = K
        g1[6] = 0;                                // stride0 hi | stride1 lo
        g1[7] = 0;                                // stride1 hi

        v4i g2 = {};                              // unused (2-D tensor)
        v4i g3 = {};
        v8i g4 = {};                              // extra group (clang-23 form)

        __builtin_amdgcn_tensor_load_to_lds(g0, g1, g2, g3, g4, /*cpol=*/0);
        __builtin_amdgcn_s_wait_tensorcnt(0);     // s_wait_tensorcnt 0x0
    }
    __syncthreads();                              // LDS now holds W

    const int lane = threadIdx.x & 31;
    const int wave = threadIdx.x >> 5;
    const int tm   = blockIdx.x * 8 + wave;       // 16-row tile index
    if (tm * 16 >= M) return;                     // wave-uniform (after barrier)

    const int half = lane >> 4;                   // 0: lanes 0-15, 1: lanes 16-31
    const int l16  = lane & 15;
    const float* Arow = A + (size_t)(tm * 16 + l16) * lda;

    v8f acc[8] = {};
    for (int k0 = 0; k0 < K; k0 += 4) {
        v2f a;                                    // A 16x4: v0=K{k0|k0+2}, v1=+1
        a.x = Arow[k0 + 2 * half];
        a.y = Arow[k0 + 2 * half + 1];

        v2f bf[8];                                // preload all 8 B fragments
#pragma unroll
        for (int tn = 0; tn < 8; ++tn) {          // B 4x16 from LDS, mirrored
            const float* Wr = ldsW + (size_t)(tn * 16 + l16) * K;
            bf[tn].x = Wr[k0 + 2 * half];
            bf[tn].y = Wr[k0 + 2 * half + 1];
        }
#pragma unroll
        for (int tn = 0; tn < 8; ++tn) {
            acc[tn] = __builtin_amdgcn_wmma_f32_16x16x4_f32(
                          false, a, false, bf[tn], (short)0, acc[tn],
                          false, false);
        }
    }

#pragma unroll
    for (int tn = 0; tn < 8; ++tn) {
        const int col = tn * 16 + l16;
        const float bv = bias ? bias[col] : 0.0f;
#pragma unroll
        for (int r = 0; r < 8; ++r) {             // C/D: M = r + 8*half
            const int row = tm * 16 + r + 8 * half;
            float v = acc[tn][r] + bv;
            if (doRelu) v = fmaxf(v, 0.0f);
            out[(size_t)row * ldo + col] = v;
        }
    }
}

// ---------------------------------------------------------------------------
// Ordered-uint encoding so float segment-max can use u32 atomicMax.
// ---------------------------------------------------------------------------
__device__ __forceinline__ unsigned f2ord(float f) {
    unsigned u = __float_as_uint(f);
    return (u & 0x80000000u) ? ~u : (u | 0x80000000u);
}
__device__ __forceinline__ float ord2f(unsigned u) {
    return (u & 0x80000000u) ? __uint_as_float(u & 0x7FFFFFFFu)
                             : __uint_as_float(~u);
}

__global__ void init_accum_kernel(float* __restrict__ rst,
                                  unsigned* __restrict__ smax,
                                  float* __restrict__ denom,
                                  int NH, int NHd) {
    const int i = blockIdx.x * blockDim.x + threadIdx.x;
    if (i < NH)  rst[i] = 0.0f;
    if (i < NHd) { smax[i] = 0u; denom[i] = 0.0f; }  // 0 < f2ord(any real score)
}

// One wave per edge: per-head attention logits + segment max (L2 atomics).
__global__ void edge_score_kernel(const float* __restrict__ hsrc,
                                  const float* __restrict__ hdst,
                                  const int* __restrict__ src,
                                  const int* __restrict__ dst,
                                  const float* __restrict__ attn,
                                  float* __restrict__ score,
                                  unsigned* __restrict__ smax, int E) {
    const int gid  = blockIdx.x * blockDim.x + threadIdx.x;
    const int e    = gid >> 5;
    const int lane = gid & 31;
    if (e >= E) return;
    const int s = src[e], d = dst[e];

    const float4 hs = *(const float4*)(hsrc + (size_t)s * HID + lane * 4);
    const float4 hd = *(const float4*)(hdst + (size_t)d * HID + lane * 4);
    const float4 at = *(const float4*)(attn + lane * 4);

    float p = 0.0f, v;
    v = hs.x + hd.x; v = (v >= 0.0f) ? v : NEG_SLOPE * v; p += at.x * v;
    v = hs.y + hd.y; v = (v >= 0.0f) ? v : NEG_SLOPE * v; p += at.y * v;
    v = hs.z + hd.z; v = (v >= 0.0f) ? v : NEG_SLOPE * v; p += at.z * v;
    v = hs.w + hd.w; v = (v >= 0.0f) ? v : NEG_SLOPE * v; p += at.w * v;

#pragma unroll
    for (int off = 1; off < 8; off <<= 1) p += __shfl_xor(p, off, 32);

    if ((lane & 7) == 0) {
        const int h = lane >> 3;
        score[(size_t)e * H_HEADS + h] = p;
        atomicMax(&smax[(size_t)d * H_HEADS + h], f2ord(p));
    }
}

// One thread per (edge, head): ex = exp(score - max); denom += ex (in place).
__global__ void edge_exp_kernel(const int* __restrict__ dst,
                                const unsigned* __restrict__ smax,
                                float* __restrict__ score,
                                float* __restrict__ denom, int EH) {
    const int i = blockIdx.x * blockDim.x + threadIdx.x;
    if (i >= EH) return;
    const int e = i >> 2, h = i & 3;
    const int d = dst[e];
    const float m  = ord2f(smax[(size_t)d * H_HEADS + h]);
    const float ex = __expf(score[i] - m);
    score[i] = ex;
    atomicAdd(&denom[(size_t)d * H_HEADS + h], ex);
}

// One wave per edge: rst[dst] += h_src[src] * ex  (unnormalized; L2 atomics).
__global__ void edge_accum_kernel(const float* __restrict__ hsrc,
                                  const int* __restrict__ src,
                                  const int* __restrict__ dst,
                                  const float* __restrict__ ex,
                                  float* __restrict__ rst, int E) {
    const int gid  = blockIdx.x * blockDim.x + threadIdx.x;
    const int e    = gid >> 5;
    const int lane = gid & 31;
    if (e >= E) return;
    const int s = src[e], d = dst[e];
    const int h = lane >> 3;
    const float a = ex[(size_t)e * H_HEADS + h];
    const float4 hs = *(const float4*)(hsrc + (size_t)s * HID + lane * 4);
    float* rp = rst + (size_t)d * HID + lane * 4;
    atomicAdd(rp + 0, hs.x * a);
    atomicAdd(rp + 1, hs.y * a);
    atomicAdd(rp + 2, hs.z * a);
    atomicAdd(rp + 3, hs.w * a);
}

// Per (node, feature): stacked[.,rel] = relu(rst/denom + res) — res already there.
__global__ void finalize_kernel(const float* __restrict__ rst,
                                const float* __restrict__ denom,
                                float* __restrict__ stackedCol, int total) {
    const int i = blockIdx.x * blockDim.x + threadIdx.x;
    if (i >= total) return;
    const int n = i >> 7, j = i & 127;
    const int h = j >> 5;
    const float den = denom[(size_t)n * H_HEADS + h];
    const float v = (den > 0.0f) ? rst[i] / den : 0.0f;   // empty segment -> 0
    float* sp = stackedCol + (size_t)n * (2 * HID) + j;
    *sp = fmaxf(v + *sp, 0.0f);
}

extern "C" void kernel_launch(void* const* d_in, const int* in_sizes, int n_in,
                              void* d_out, int out_size, void* d_ws, size_t ws_size,
                              hipStream_t stream) {
    const float* feat_agent = (const float*)d_in[0];
    const float* feat_user  = (const float*)d_in[1];
    const float* feat_power = (const float*)d_in[2];
    const int*   src_r[2]   = { (const int*)d_in[3], (const int*)d_in[5] };
    const int*   dst_r[2]   = { (const int*)d_in[4], (const int*)d_in[6] };
    const float* feat_src_r[2] = { feat_user, feat_power };
    const float* W_aggr = (const float*)d_in[21];
    const float* b_aggr = (const float*)d_in[22];

    // workspace layout (reused across the two relations, processed serially)
    float* ws = (float*)d_ws;
    size_t off = 0;
    float*    h_src  = ws + off; off += (size_t)N_SRC * HID;
    float*    h_dst  = ws + off; off += (size_t)N_AGENT * HID;
    float*    rst    = ws + off; off += (size_t)N_AGENT * HID;
    float*    score  = ws + off; off += (size_t)E_EDGE * H_HEADS;
    unsigned* smax   = (unsigned*)(ws + off); off += (size_t)N_AGENT * H_HEADS;
    float*    denom  = ws + off; off += (size_t)N_AGENT * H_HEADS;
    float*    stacked = ws + off;                                  // N x 256

    const dim3 gemmBlk(256, 1, 1);
    const dim3 gSrc((N_SRC + 127) / 128, 1, 1);
    const dim3 gDst((N_AGENT + 127) / 128, 1, 1);
    const size_t smem16  = (size_t)HID * F_SRC   * sizeof(float);  //   8 KB
    const size_t smem32  = (size_t)HID * F_AGENT * sizeof(float);  //  16 KB
    const size_t smem256 = (size_t)HID * 2 * HID * sizeof(float);  // 128 KB

    for (int rel = 0; rel < 2; ++rel) {
        const int pb = 7 + rel * 7;
        const float* W_src = (const float*)d_in[pb + 0];
        const float* b_src = (const float*)d_in[pb + 1];
        const float* W_dst = (const float*)d_in[pb + 2];
        const float* b_dst = (const float*)d_in[pb + 3];
        const float* attn  = (const float*)d_in[pb + 4];
        const float* W_res = (const float*)d_in[pb + 5];
        const float* b_res = (const float*)d_in[pb + 6];

        init_accum_kernel<<<(N_AGENT * HID + 255) / 256, 256, 0, stream>>>(
            rst, smax, denom, N_AGENT * HID, N_AGENT * H_HEADS);

        // projections: TDM-staged weights + fp32 WMMA
        wmma_gemm_tdm_kernel<<<gSrc, gemmBlk, smem16, stream>>>(
            feat_src_r[rel], F_SRC, W_src, b_src, h_src, HID, N_SRC, F_SRC, 0);
        wmma_gemm_tdm_kernel<<<gDst, gemmBlk, smem32, stream>>>(
            feat_agent, F_AGENT, W_dst, b_dst, h_dst, HID, N_AGENT, F_AGENT, 0);
        wmma_gemm_tdm_kernel<<<gDst, gemmBlk, smem32, stream>>>(
            feat_agent, F_AGENT, W_res, b_res, stacked + rel * HID, 2 * HID,
            N_AGENT, F_AGENT, 0);

        // edge phase (L2-resident gathers + atomics)
        edge_score_kernel<<<((size_t)E_EDGE * 32 + 255) / 256, 256, 0, stream>>>(
            h_src, h_dst, src_r[rel], dst_r[rel], attn, score, smax, E_EDGE);
        edge_exp_kernel<<<(E_EDGE * H_HEADS + 255) / 256, 256, 0, stream>>>(
            dst_r[rel], smax, score, denom, E_EDGE * H_HEADS);
        edge_accum_kernel<<<((size_t)E_EDGE * 32 + 255) / 256, 256, 0, stream>>>(
            h_src, src_r[rel], dst_r[rel], score, rst, E_EDGE);
        finalize_kernel<<<(N_AGENT * HID + 255) / 256, 256, 0, stream>>>(
            rst, denom, stacked + rel * HID, N_AGENT * HID);
    }

    // aggregation: out = relu(stacked @ W_aggr^T + b_aggr), K = 256, via WMMA
    wmma_gemm_tdm_kernel<<<gDst, gemmBlk, smem256, stream>>>(
        stacked, 2 * HID, W_aggr, b_aggr, (float*)d_out, HID,
        N_AGENT, 2 * HID, 1);
}